// SparseMaxPoolTestTorch_45311904972973
// MI455X (gfx1250) — compile-verified
//
#include <hip/hip_runtime.h>
#include <hip/hip_bf16.h>

// ---------------------------------------------------------------------------
// SparseMaxPool3d x2 (K=3,S=2,P=1) on (2,96,128,128,32), fused first layer.
//
// Stage 1: zero padded layer-1 grid (2,50,66,66,32) uint32 in d_ws (sentinel 0)
// Stage 2: scatter points -> layer-1 output sites directly (<=8 sites/point)
//          via atomicMax(uint) on order-preserving float encoding
// Stage 3: dense 3^3/s2 max-pool of layer-1 grid -> (2,24,32,32,32) f32.
//          Input slab staged into LDS with TENSOR_LOAD_TO_LDS (TDM), output
//          tile written back with TENSOR_STORE_FROM_LDS.
// ---------------------------------------------------------------------------

#define L1D 48
#define L1H 64
#define L1W 64
#define CH  32

// padded layer-1 grid dims (halo of 1 on each side, sentinel 0)
#define PZ 50
#define PY 66
#define PX 66
#define STRIDE_X  (CH)                 // 32
#define STRIDE_Y  (PX * CH)            // 2112
#define STRIDE_Z  (PY * PX * CH)       // 139392
#define STRIDE_B  (PZ * PY * PX * CH)  // 6969600
#define WS_WORDS  (2 * STRIDE_B)       // 13939200 uints (~53.2 MB)

// output dims
#define OD 24
#define OH 32
#define OW 32
#define OSTR_X (CH)                    // 32
#define OSTR_Y (OW * CH)               // 1024
#define OSTR_Z (OH * OW * CH)          // 32768
#define OSTR_B (OD * OH * OW * CH)     // 786432

// layer-2 tile: 2x4x4 outputs -> 5x9x9 input slab (x32 channels)
#define TZ 5
#define TY 9
#define TX 9
#define TILE_WORDS (TZ * TY * TX * CH) // 12960 (51840 B LDS)
#define OTILE_WORDS (2 * 4 * 4 * CH)   // 1024  (4096 B LDS)

typedef unsigned int u32;
typedef unsigned long long u64;
typedef u32 uint32x4 __attribute__((ext_vector_type(4)));
typedef int int32x4  __attribute__((ext_vector_type(4)));
typedef int int32x8  __attribute__((ext_vector_type(8)));

#if defined(__has_builtin)
#if __has_builtin(__builtin_amdgcn_tensor_load_to_lds)
#define HAVE_TDM_LD 1
#endif
#if __has_builtin(__builtin_amdgcn_tensor_store_from_lds)
#define HAVE_TDM_ST 1
#endif
#endif

__device__ __forceinline__ u32 enc_f32(float f) {
  u32 u = __float_as_uint(f);
  return (u & 0x80000000u) ? ~u : (u | 0x80000000u);
}
__device__ __forceinline__ float dec_f32(u32 u) {
  u32 b = (u & 0x80000000u) ? (u & 0x7FFFFFFFu) : ~u;
  return __uint_as_float(b);
}

// ---------------- Stage 1: zero workspace ----------------
__global__ void zero_ws_kernel(u32* __restrict__ p, int n4) {
  int t = blockIdx.x * blockDim.x + threadIdx.x;
  if (t < n4) {
    ((uint4*)p)[t] = make_uint4(0u, 0u, 0u, 0u);
  }
}

// ---------------- Stage 2: fused densify + pool layer 1 ----------------
// Thread t handles point i = t>>5, channel c = t&31 (one wave per point).
__global__ void scatter_pool1_kernel(const float* __restrict__ feat,
                                     const int* __restrict__ coors,
                                     u32* __restrict__ l1, int n_pts) {
  int t = blockIdx.x * blockDim.x + threadIdx.x;
  if (t >= n_pts * CH) return;
  int i = t >> 5;
  int c = t & 31;
  int b = coors[4 * i + 0];
  int z = coors[4 * i + 1];
  int y = coors[4 * i + 2];
  int x = coors[4 * i + 3];
  u32 v = enc_f32(feat[t]);

  int za[2], ya[2], xa[2];
  int nz = 1, ny = 1, nx = 1;
  za[0] = z >> 1;
  if ((z & 1) && ((z >> 1) + 1) < L1D) za[nz++] = (z >> 1) + 1;
  ya[0] = y >> 1;
  if ((y & 1) && ((y >> 1) + 1) < L1H) ya[ny++] = (y >> 1) + 1;
  xa[0] = x >> 1;
  if ((x & 1) && ((x >> 1) + 1) < L1W) xa[nx++] = (x >> 1) + 1;

  long bb = (long)b * STRIDE_B;
  for (int zi = 0; zi < nz; ++zi)
    for (int yi = 0; yi < ny; ++yi)
      for (int xi = 0; xi < nx; ++xi) {
        long idx = bb + (long)(za[zi] + 1) * STRIDE_Z +
                   (long)(ya[yi] + 1) * STRIDE_Y +
                   (long)(xa[xi] + 1) * STRIDE_X + c;
        atomicMax(&l1[idx], v);
      }
}

// ---------------- Stage 3: pool layer 2, TDM in and TDM out ----------------
__global__ __launch_bounds__(256) void pool2_kernel(const u32* __restrict__ l1,
                                                    float* __restrict__ out) {
  __shared__ u32 tile[TILE_WORDS];     // input slab  [z:5][y:9][x:9][c:32]
#if defined(HAVE_TDM_ST)
  __shared__ float otile[OTILE_WORDS]; // output tile [z:2][y:4][x:4][c:32]
#endif

  int xb = blockIdx.x;            // 0..7  -> ox0 = 4*xb
  int yb = blockIdx.y;            // 0..7  -> oy0 = 4*yb
  int b  = blockIdx.z / 12;       // 0..1
  int zb = blockIdx.z % 12;       // 0..11 -> oz0 = 2*zb
  int oz0 = zb * 2, oy0 = yb * 4, ox0 = xb * 4;

  // padded-grid origin of the input slab
  long base_elem = (long)b * STRIDE_B + (long)(2 * oz0) * STRIDE_Z +
                   (long)(2 * oy0) * STRIDE_Y + (long)(2 * ox0) * STRIDE_X;

#if defined(HAVE_TDM_LD)
  if (threadIdx.x < 32) { // one wave issues the DMA (EXEC ignored by TDM)
    u64 ga = (u64)(size_t)(l1 + base_elem);
    u32 lds_addr = (u32)(size_t)(&tile[0]); // low 32 bits of flat = LDS addr

    uint32x4 g0;
    g0[0] = 1u;                 // count=1, user descriptor, gather off
    g0[1] = lds_addr;           // lds_addr
    g0[2] = (u32)ga;            // global_addr[31:0]
    g0[3] = ((u32)(ga >> 32) & 0x01FFFFFFu) | (2u << 30); // addr[56:32]|type=2

    int32x8 g1;
    g1[0] = (int)(2u << 16);            // data_size=4B; mask=0; no pad/iter
    g1[1] = (int)(32u << 16);           // tensor_dim0 lo16 = 32 (channels)
    g1[2] = (int)((u32)PX << 16);       // dim0 hi=0 | tensor_dim1 lo16 = 66
    g1[3] = (int)(32u << 16);           // dim1 hi=0 | tile_dim0 = 32
    g1[4] = (int)((u32)TX | ((u32)TY << 16)); // tile_dim1=9 | tile_dim2=9
    g1[5] = (int)STRIDE_X;              // tensor_dim0_stride lo32 = 32
    g1[6] = (int)((u32)STRIDE_Y << 16); // str0 hi=0 | tensor_dim1_stride lo16
    g1[7] = 0;                          // tensor_dim1_stride hi32 = 0

    int32x4 g2;
    g2[0] = PY;                         // tensor_dim2 = 66
    g2[1] = PZ;                         // tensor_dim3 = 50
    g2[2] = (int)STRIDE_Z;              // tensor_dim2_stride lo32 = 139392
    g2[3] = (int)((u32)TZ << 16);       // str2 hi=0 | tile_dim3 = 5

    int32x4 g3;
    g3[0] = (int)STRIDE_B;              // tensor_dim3_stride lo32
    g3[1] = (int)(2u << 16);            // str3 hi=0 | tensor_dim4 lo16 = 2
    g3[2] = 0;                          // dim4 hi | tile_dim4 = 0 (unused)
    g3[3] = 0;

#if __has_include(<hip/amd_detail/amd_gfx1250_TDM.h>)
    int32x8 gx = {0, 0, 0, 0, 0, 0, 0, 0};
    __builtin_amdgcn_tensor_load_to_lds(g0, g1, g2, g3, gx, 0);
#else
    __builtin_amdgcn_tensor_load_to_lds(g0, g1, g2, g3, 0);
#endif
    __builtin_amdgcn_s_wait_tensorcnt(0);
  }
#else
  // Fallback: cooperative plain loads into the same LDS layout.
  for (int e = threadIdx.x; e < TILE_WORDS; e += 256) {
    int c = e & 31;
    int r = e >> 5;
    int x = r % TX; r /= TX;
    int y = r % TY;
    int z = r / TY;
    tile[e] = l1[base_elem + (long)z * STRIDE_Z + (long)y * STRIDE_Y +
                 (long)x * STRIDE_X + c];
  }
#endif
  __syncthreads();

  // 1024 outputs per block, 4 per thread; lanes = consecutive channels
  // (bank-conflict free: within a wave the spatial site is uniform).
  for (int o = threadIdx.x; o < OTILE_WORDS; o += 256) {
    int c  = o & 31;
    int lx = (o >> 5) & 3;
    int ly = (o >> 7) & 3;
    int lz = (o >> 9) & 1;
    int base = ((2 * lz) * (TY * TX) + (2 * ly) * TX + (2 * lx)) * CH + c;
    u32 m = 0u;
#pragma unroll
    for (int dz = 0; dz < 3; ++dz)
#pragma unroll
      for (int dy = 0; dy < 3; ++dy)
#pragma unroll
        for (int dx = 0; dx < 3; ++dx) {
          u32 v = tile[base + (dz * (TY * TX) + dy * TX + dx) * CH];
          m = (v > m) ? v : m;
        }
    float r = (m == 0u) ? 0.0f : dec_f32(m);
#if defined(HAVE_TDM_ST)
    otile[o] = r; // [z][y][x][c] == TDM X->Y->Z traversal order
#else
    {
      int oz = oz0 + lz, oy = oy0 + ly, ox = ox0 + lx;
      long oidx = ((((long)b * OD + oz) * OH + oy) * OW + ox) * CH + c;
      out[oidx] = r;
    }
#endif
  }

#if defined(HAVE_TDM_ST)
  __syncthreads(); // all ds_stores of otile visible before DMA-out
  if (threadIdx.x < 32) {
    long obase = (long)b * OSTR_B + (long)oz0 * OSTR_Z + (long)oy0 * OSTR_Y +
                 (long)ox0 * OSTR_X;
    u64 ga = (u64)(size_t)(out + obase);
    u32 lds_addr = (u32)(size_t)(&otile[0]);

    uint32x4 g0;
    g0[0] = 1u;
    g0[1] = lds_addr;
    g0[2] = (u32)ga;
    g0[3] = ((u32)(ga >> 32) & 0x01FFFFFFu) | (2u << 30);

    int32x8 g1;
    g1[0] = (int)(2u << 16);            // data_size=4B
    g1[1] = (int)(32u << 16);           // tensor_dim0 = 32 (channels)
    g1[2] = (int)((u32)OW << 16);       // tensor_dim1 lo16 = 32 (x)
    g1[3] = (int)(32u << 16);           // tile_dim0 = 32
    g1[4] = (int)(4u | (4u << 16));     // tile_dim1=4 (x) | tile_dim2=4 (y)
    g1[5] = (int)OSTR_X;                // tensor_dim0_stride = 32
    g1[6] = (int)((u32)OSTR_Y << 16);   // tensor_dim1_stride lo16 = 1024
    g1[7] = 0;

    int32x4 g2;
    g2[0] = OH;                         // tensor_dim2 = 32 (y)
    g2[1] = OD;                         // tensor_dim3 = 24 (z)
    g2[2] = (int)OSTR_Z;                // tensor_dim2_stride = 32768
    g2[3] = (int)(2u << 16);            // tile_dim3 = 2 (z)

    int32x4 g3;
    g3[0] = (int)OSTR_B;                // tensor_dim3_stride = 786432
    g3[1] = (int)(2u << 16);            // tensor_dim4 = 2 (batch)
    g3[2] = 0;                          // tile_dim4 = 0 (unused)
    g3[3] = 0;

#if __has_include(<hip/amd_detail/amd_gfx1250_TDM.h>)
    int32x8 gx = {0, 0, 0, 0, 0, 0, 0, 0};
    __builtin_amdgcn_tensor_store_from_lds(g0, g1, g2, g3, gx, 0);
#else
    __builtin_amdgcn_tensor_store_from_lds(g0, g1, g2, g3, 0);
#endif
    __builtin_amdgcn_s_wait_tensorcnt(0);
  }
#endif
}

// ---------------------------------------------------------------------------
extern "C" void kernel_launch(void* const* d_in, const int* in_sizes, int n_in,
                              void* d_out, int out_size, void* d_ws,
                              size_t ws_size, hipStream_t stream) {
  const float* feat = (const float*)d_in[0];
  const int* coors  = (const int*)d_in[1];
  float* out        = (float*)d_out;
  u32* l1           = (u32*)d_ws;

  (void)n_in; (void)out_size;
  // workspace requirement: WS_WORDS * 4 = ~53.2 MB
  if (ws_size < (size_t)WS_WORDS * sizeof(u32)) return;

  int n_pts = in_sizes[1] / 4;

  // Stage 1: zero padded layer-1 grid (sentinel 0 == -inf in encoded domain)
  {
    int n4 = WS_WORDS / 4;
    zero_ws_kernel<<<(n4 + 255) / 256, 256, 0, stream>>>(l1, n4);
  }
  // Stage 2: fused densify + first max-pool via uint atomicMax scatter
  {
    int total = n_pts * CH;
    scatter_pool1_kernel<<<(total + 255) / 256, 256, 0, stream>>>(feat, coors,
                                                                  l1, n_pts);
  }
  // Stage 3: second max-pool, TDM-staged tiles in and out
  {
    dim3 grid(OW / 4, OH / 4, 2 * (OD / 2));
    pool2_kernel<<<grid, 256, 0, stream>>>(l1, out);
  }
}